// NeuralCRF_73186242724349
// MI455X (gfx1250) — compile-verified
//
#include <hip/hip_runtime.h>

#define B_SZ 512
#define L_SZ 2048
#define C_SZ 32
#define PF_DIST 24

typedef float v2f __attribute__((ext_vector_type(2)));
typedef float v8f __attribute__((ext_vector_type(8)));

// Swap bits 1 and 3 (involution). Used to permute the WMMA contraction
// ordering so D-fragment layout == next-step B-fragment layout.
__device__ __forceinline__ int swap13(int x) {
  return (x & ~0x0A) | ((x & 2) << 2) | ((x & 8) >> 2);
}

__device__ __forceinline__ void load_step(const float* __restrict__ eb, int t, int h,
                                          float4& a, float4& b, float4& c, float4& d) {
  const float4* p = (const float4*)(eb + (size_t)t * C_SZ + 8 * h);
  a = p[0];        // states 8h .. 8h+3    (tile 0, v=0..3)
  b = p[1];        // states 8h+4 .. 8h+7  (tile 0, v=4..7)
  c = p[4];        // states 16+8h ..      (tile 1, v=0..3)
  d = p[5];        // states 20+8h ..      (tile 1, v=4..7)
}

// ---------------------------------------------------------------------------
// Forward recursion: one wave handles 16 batches. alpha kept in exp-space with
// a carried per-batch natural-log scale; rescale every 4 steps.
// ---------------------------------------------------------------------------
__global__ __launch_bounds__(32) void crf_forward(const float* __restrict__ em,
                                                  const float* __restrict__ trans,
                                                  float* __restrict__ logZ) {
  const int lane = threadIdx.x;
  const int h = lane >> 4;          // lane half
  const int n = lane & 15;          // batch column / A-row index
  const int b = blockIdx.x * 16 + n;
  const float* eb = em + (size_t)b * L_SZ * C_SZ;

  // Constant A fragments: A[i][kappa] = exp(T[i][swap13(kappa)]).
  // A layout (16x4 f32): lane%16 = M = i_sub; VGPR0 holds K=2h, VGPR1 K=2h+1.
  v2f Af0[8], Af1[8];
#pragma unroll
  for (int kk = 0; kk < 8; ++kk) {
    int k0 = 4 * kk + 2 * h;
    int j0 = swap13(k0);                // swap13 keeps bit0, so j(k0+1)=j0+1
    Af0[kk].x = __expf(trans[(0 * 16 + n) * C_SZ + j0]);
    Af0[kk].y = __expf(trans[(0 * 16 + n) * C_SZ + j0 + 1]);
    Af1[kk].x = __expf(trans[(1 * 16 + n) * C_SZ + j0]);
    Af1[kk].y = __expf(trans[(1 * 16 + n) * C_SZ + j0 + 1]);
  }

  // alpha_exp in D layout: a0[v] = state v+8h, a1[v] = state 16+v+8h, batch n.
  float a0[8], a1[8];
  float4 i0, i1, i2, i3;
  load_step(eb, 0, h, i0, i1, i2, i3);
  {
    float im0[8] = {i0.x, i0.y, i0.z, i0.w, i1.x, i1.y, i1.z, i1.w};
    float im1[8] = {i2.x, i2.y, i2.z, i2.w, i3.x, i3.y, i3.z, i3.w};
    float m = im0[0];
#pragma unroll
    for (int v = 1; v < 8; ++v) m = fmaxf(m, im0[v]);
#pragma unroll
    for (int v = 0; v < 8; ++v) m = fmaxf(m, im1[v]);
    m = fmaxf(m, __shfl_xor(m, 16));
#pragma unroll
    for (int v = 0; v < 8; ++v) {
      a0[v] = __expf(im0[v] - m);
      a1[v] = __expf(im1[v] - m);
    }
    i0 = i1 = i2 = i3 = float4{0, 0, 0, 0};
    // carried log-scale (identical across both lane halves of a batch)
    a0[0] = a0[0]; // no-op
    // store scale in a named var:
    // (declared below to keep scope simple)
    (void)0;
  }
  float logscale;
  {
    // recompute m cheaply is wasteful; instead recover: done above — redo init
  }
  // --- redo init cleanly with logscale in scope ---
  load_step(eb, 0, h, i0, i1, i2, i3);
  {
    float im0[8] = {i0.x, i0.y, i0.z, i0.w, i1.x, i1.y, i1.z, i1.w};
    float im1[8] = {i2.x, i2.y, i2.z, i2.w, i3.x, i3.y, i3.z, i3.w};
    float m = im0[0];
#pragma unroll
    for (int v = 1; v < 8; ++v) m = fmaxf(m, im0[v]);
#pragma unroll
    for (int v = 0; v < 8; ++v) m = fmaxf(m, im1[v]);
    m = fmaxf(m, __shfl_xor(m, 16));
#pragma unroll
    for (int v = 0; v < 8; ++v) {
      a0[v] = __expf(im0[v] - m);
      a1[v] = __expf(im1[v] - m);
    }
    logscale = m;
  }

  // 1-deep register pipeline for emissions
  float4 n0, n1, n2, n3;
  load_step(eb, 1, h, n0, n1, n2, n3);

#pragma unroll 1
  for (int t = 1; t < L_SZ; ++t) {
    float4 c0 = n0, c1 = n1, c2 = n2, c3 = n3;
    if (t + 1 < L_SZ) load_step(eb, t + 1, h, n0, n1, n2, n3);
    if (t + PF_DIST < L_SZ)
      __builtin_prefetch(eb + (size_t)(t + PF_DIST) * C_SZ, 0, 1);

    // s = alpha_exp @ exp(T)^T  via 2 output tiles x 8 chained K=4 WMMAs.
    // B fragments are direct aliases of acc registers (swap13 ordering).
    v8f s0 = {0.f, 0.f, 0.f, 0.f, 0.f, 0.f, 0.f, 0.f};
    v8f s1 = {0.f, 0.f, 0.f, 0.f, 0.f, 0.f, 0.f, 0.f};
#pragma unroll
    for (int kk = 0; kk < 8; ++kk) {
      const int vs = 4 * (kk & 1) + 2 * ((kk >> 1) & 1);
      v2f bf;
      bf.x = (kk < 4) ? a0[vs] : a1[vs];
      bf.y = (kk < 4) ? a0[vs + 1] : a1[vs + 1];
      s0 = __builtin_amdgcn_wmma_f32_16x16x4_f32(false, Af0[kk], false, bf,
                                                 (short)0, s0, false, false);
      s1 = __builtin_amdgcn_wmma_f32_16x16x4_f32(false, Af1[kk], false, bf,
                                                 (short)0, s1, false, false);
    }

    // alpha_exp = s * exp(e_t)
    float em0[8] = {c0.x, c0.y, c0.z, c0.w, c1.x, c1.y, c1.z, c1.w};
    float em1[8] = {c2.x, c2.y, c2.z, c2.w, c3.x, c3.y, c3.z, c3.w};
#pragma unroll
    for (int v = 0; v < 8; ++v) {
      a0[v] = s0[v] * __expf(em0[v]);
      a1[v] = s1[v] * __expf(em1[v]);
    }

    // rescale every 4 steps (uniform branch; growth bounded well inside fp32)
    if ((t & 3) == 3) {
      float m = a0[0];
#pragma unroll
      for (int v = 1; v < 8; ++v) m = fmaxf(m, a0[v]);
#pragma unroll
      for (int v = 0; v < 8; ++v) m = fmaxf(m, a1[v]);
      m = fmaxf(m, __shfl_xor(m, 16));
      float r = 1.0f / m;
#pragma unroll
      for (int v = 0; v < 8; ++v) {
        a0[v] *= r;
        a1[v] *= r;
      }
      logscale += __logf(m);
    }
  }

  float ssum = 0.f;
#pragma unroll
  for (int v = 0; v < 8; ++v) ssum += a0[v] + a1[v];
  ssum += __shfl_xor(ssum, 16);
  float lz = logscale + __logf(ssum);
  if (h == 0) logZ[b] = lz;
}

// ---------------------------------------------------------------------------
// Path score: one block per batch. Also warms L2 with the full emissions
// tensor (every (b,t) 128B line is touched) before the recursion kernel runs.
// ---------------------------------------------------------------------------
__global__ __launch_bounds__(256) void crf_score(const float* __restrict__ em,
                                                 const float* __restrict__ trans,
                                                 const int* __restrict__ tags,
                                                 const int* __restrict__ mask,
                                                 float* __restrict__ score_out) {
  __shared__ float Tsh[C_SZ * C_SZ];
  __shared__ float redf[256];
  __shared__ int redi[256];
  const int b = blockIdx.x, tid = threadIdx.x;
  for (int i = tid; i < C_SZ * C_SZ; i += 256) Tsh[i] = trans[i];
  __syncthreads();

  const int* tg = tags + (size_t)b * L_SZ;
  const int* mk = mask + (size_t)b * L_SZ;
  const float* eb = em + (size_t)b * L_SZ * C_SZ;

  float acc = 0.f;
  int msum = 0;
  for (int t = tid; t < L_SZ; t += 256) {
    int tag = tg[t];
    int m = mk[t];
    msum += m;
    if (t < L_SZ - 1) {
      int tagn = tg[t + 1];
      int mn = mk[t + 1];
      acc += eb[(size_t)t * C_SZ + tag] * (float)m +
             Tsh[tag * C_SZ + tagn] * (float)mn;
    }
  }
  redf[tid] = acc;
  redi[tid] = msum;
  __syncthreads();
  for (int s = 128; s > 0; s >>= 1) {
    if (tid < s) {
      redf[tid] += redf[tid + s];
      redi[tid] += redi[tid + s];
    }
    __syncthreads();
  }
  if (tid == 0) {
    int ms = redi[0];
    int last_idx = ms - 1;
    if (last_idx < 0) last_idx = 0;
    int last_tag = tg[last_idx];
    int le_t = (ms < 1 ? 1 : ms) - 1;
    score_out[b] = redf[0] + eb[(size_t)le_t * C_SZ + last_tag];
  }
}

// ---------------------------------------------------------------------------
// mean(logZ - score), deterministic single-block tree reduction.
// ---------------------------------------------------------------------------
__global__ __launch_bounds__(512) void crf_final(const float* __restrict__ score,
                                                 const float* __restrict__ logZ,
                                                 float* __restrict__ out) {
  __shared__ float red[512];
  const int tid = threadIdx.x;
  red[tid] = logZ[tid] - score[tid];
  __syncthreads();
  for (int s = 256; s > 0; s >>= 1) {
    if (tid < s) red[tid] += red[tid + s];
    __syncthreads();
  }
  if (tid == 0) out[0] = red[0] * (1.0f / (float)B_SZ);
}

extern "C" void kernel_launch(void* const* d_in, const int* in_sizes, int n_in,
                              void* d_out, int out_size, void* d_ws, size_t ws_size,
                              hipStream_t stream) {
  const float* emissions = (const float*)d_in[0];
  const float* transitions = (const float*)d_in[1];
  const int* tags = (const int*)d_in[2];
  const int* mask = (const int*)d_in[3];
  float* out = (float*)d_out;

  float* score = (float*)d_ws;          // B_SZ floats
  float* logZ = score + B_SZ;           // B_SZ floats

  // 1) score first: also pulls all of emissions (128 MiB) into the 192 MB L2
  crf_score<<<B_SZ, 256, 0, stream>>>(emissions, transitions, tags, mask, score);
  // 2) sequential forward recursion, 16 batches per wave, 32 waves
  crf_forward<<<B_SZ / 16, 32, 0, stream>>>(emissions, transitions, logZ);
  // 3) deterministic mean
  crf_final<<<1, B_SZ, 0, stream>>>(score, logZ, out);
}